// SE_Block_3616362463724
// MI455X (gfx1250) — compile-verified
//
#include <hip/hip_runtime.h>
#include <hip/hip_bf16.h>
#include <math.h>

// ---------------------------------------------------------------------------
// SE block: pool -> MLP(WMMA f32 16x16x4) -> top-k mask -> scale
// x: [64,256,56,56] f32.  Bandwidth-bound: ~615MB traffic -> ~26us @ 23.3TB/s.
// ---------------------------------------------------------------------------

typedef __attribute__((ext_vector_type(2))) float v2f;
typedef __attribute__((ext_vector_type(8))) float v8f;

#define BATCH   64
#define CH      256
#define HW      3136            // 56*56
#define HW4     (HW / 4)        // 784 float4s per (b,c) row
#define RED     16              // CH / r
#define KEEP_IDX 204            // position-1, position = round(0.8*256) = 205

// ---------------------------------------------------------------------------
// Kernel 1: global average pool.  One 256-thread block per (b,c) row of 3136
// contiguous floats; float4 coalesced loads, LDS tree reduction.
// ---------------------------------------------------------------------------
__global__ void __launch_bounds__(256) se_pool_kernel(const float* __restrict__ x,
                                                      float* __restrict__ y) {
    __shared__ float red[256];
    const int bc  = blockIdx.x;
    const int tid = threadIdx.x;
    const float4* __restrict__ x4 = (const float4*)(x + (size_t)bc * HW);
    float sum = 0.0f;
    for (int i = tid; i < HW4; i += 256) {
        float4 v = x4[i];
        sum += (v.x + v.y) + (v.z + v.w);
    }
    red[tid] = sum;
    __syncthreads();
    #pragma unroll
    for (int s = 128; s > 0; s >>= 1) {
        if (tid < s) red[tid] += red[tid + s];
        __syncthreads();
    }
    if (tid == 0) y[bc] = red[0] * (1.0f / (float)HW);
}

// ---------------------------------------------------------------------------
// Kernel 2: squeeze-excite MLP on the 64x256 pooled matrix using
// V_WMMA_F32_16X16X4_F32.  One block, 128 threads = 4 wave32s; wave w owns
// the 16-row tile m0 = 16*w.
//
// f32 A 16x4 layout: lane L -> M = L&15; VGPR0/1 = K {0,1} (lanes 0-15) or
// {2,3} (lanes 16-31)  =>  per-lane contiguous float2 at column k + 2*(L>>4).
// B 4x16 mirrors with N striped across lanes.  C/D: VGPR i -> rows i / i+8.
// ---------------------------------------------------------------------------
__global__ void __launch_bounds__(128) se_mlp_wmma_kernel(
    const float* __restrict__ y,    // [64,256]
    const float* __restrict__ W1,   // [16,256]   z = relu(y @ W1^T)
    const float* __restrict__ W2,   // [256,16]   s = sigmoid(z @ W2^T)
    float* __restrict__ s)          // [64,256]
{
    __shared__ float zsh[BATCH * RED];         // 64x16 hidden activations
    const int tid  = threadIdx.x;
    const int lane = tid & 31;
    const int wave = tid >> 5;                 // 0..3
    const int m0   = wave * 16;
    const int half = lane >> 4;                // which K-pair / row-half
    const int l16  = lane & 15;
    const int mA   = m0 + l16;                 // A-fragment row for this lane

    // ---- GEMM1: [64,256] x [256,16] -> relu -> zsh ----
    v8f acc = {};
    for (int k = 0; k < CH; k += 4) {
        v2f a = *(const v2f*)(y  + mA  * CH + k + 2 * half);
        // B[k][n] = W1[n][k]  (n = l16): contiguous pair along W1's row
        v2f b = *(const v2f*)(W1 + l16 * CH + k + 2 * half);
        acc = __builtin_amdgcn_wmma_f32_16x16x4_f32(
                  false, a, false, b, (short)0, acc, false, false);
    }
    #pragma unroll
    for (int i = 0; i < 8; ++i) {
        const int m = m0 + i + 8 * half;
        zsh[m * RED + l16] = fmaxf(acc[i], 0.0f);
    }
    __syncthreads();

    // ---- GEMM2: [64,16] x [16,256] -> sigmoid -> s ----
    for (int nt = 0; nt < CH / 16; ++nt) {
        const int n0 = nt * 16;
        v8f acc2 = {};
        #pragma unroll
        for (int k = 0; k < RED; k += 4) {
            v2f a;
            a.x = zsh[mA * RED + k + 2 * half];
            a.y = zsh[mA * RED + k + 2 * half + 1];
            // B[k][n] = W2[n][k]: contiguous pair along W2's (length-16) row
            v2f b = *(const v2f*)(W2 + (n0 + l16) * RED + k + 2 * half);
            acc2 = __builtin_amdgcn_wmma_f32_16x16x4_f32(
                       false, a, false, b, (short)0, acc2, false, false);
        }
        #pragma unroll
        for (int i = 0; i < 8; ++i) {
            const int m = m0 + i + 8 * half;
            const float v = acc2[i];
            s[m * CH + n0 + l16] = 1.0f / (1.0f + __expf(-v));
        }
    }
}

// ---------------------------------------------------------------------------
// Kernel 3: per-row order statistic.  thresh = sorted(row)[204]; an element v
// equals sorted[204] iff  (#<v) <= 204 < (#<=v).  mask = (v <= thresh).
// 256^2 compares per row x 64 rows: negligible.
// ---------------------------------------------------------------------------
__global__ void __launch_bounds__(256) se_mask_kernel(const float* __restrict__ s,
                                                      float* __restrict__ mask) {
    __shared__ float row[CH];
    __shared__ float thresh;
    const int b = blockIdx.x;
    const int t = threadIdx.x;
    const float v = s[b * CH + t];
    row[t] = v;
    __syncthreads();
    int lt = 0, le = 0;
    for (int j = 0; j < CH; ++j) {
        const float r = row[j];
        lt += (r <  v) ? 1 : 0;
        le += (r <= v) ? 1 : 0;
    }
    if (lt <= KEEP_IDX && KEEP_IDX < le) thresh = v;   // all writers agree on v
    __syncthreads();
    mask[b * CH + t] = (v <= thresh) ? 1.0f : 0.0f;
}

// ---------------------------------------------------------------------------
// Kernel 4: streaming scale.  One block per (b,c) row; the per-row scalar
// mask is a single SMEM-cached load, then pure float4 stream: the kernel that
// sets the 23.3 TB/s roofline.  global_prefetch to cover latency.
// ---------------------------------------------------------------------------
__global__ void __launch_bounds__(256) se_scale_kernel(const float* __restrict__ x,
                                                       const float* __restrict__ mask,
                                                       float* __restrict__ out) {
    const int bc  = blockIdx.x;
    const int tid = threadIdx.x;
    const float m = mask[bc];
    const float4* __restrict__ x4 = (const float4*)(x  + (size_t)bc * HW);
    float4* __restrict__       o4 = (float4*)(out + (size_t)bc * HW);
    for (int i = tid; i < HW4; i += 256) {
        if (i + 256 < HW4) __builtin_prefetch(&x4[i + 256], 0, 0);
        float4 v = x4[i];
        v.x *= m; v.y *= m; v.z *= m; v.w *= m;
        o4[i] = v;
    }
}

// ---------------------------------------------------------------------------
extern "C" void kernel_launch(void* const* d_in, const int* in_sizes, int n_in,
                              void* d_out, int out_size, void* d_ws, size_t ws_size,
                              hipStream_t stream) {
    (void)in_sizes; (void)n_in; (void)out_size; (void)ws_size;
    const float* x  = (const float*)d_in[0];   // [64,256,56,56]
    const float* W1 = (const float*)d_in[1];   // [16,256]
    const float* W2 = (const float*)d_in[2];   // [256,16]
    float* out = (float*)d_out;

    float* y    = (float*)d_ws;                // [64,256] pooled
    float* s    = y + BATCH * CH;              // [64,256] sigmoid gates
    float* mask = s + BATCH * CH;              // [64,256] 0/1 mask
    // total ws use: 3 * 16384 floats = 192 KB

    se_pool_kernel<<<BATCH * CH, 256, 0, stream>>>(x, y);
    se_mlp_wmma_kernel<<<1, 128, 0, stream>>>(y, W1, W2, s);
    se_mask_kernel<<<BATCH, CH, 0, stream>>>(s, mask);
    se_scale_kernel<<<BATCH * CH, 256, 0, stream>>>(x, mask, out);
}